// MsDL_GPU_62586263438029
// MI455X (gfx1250) — compile-verified
//
#include <hip/hip_runtime.h>
#include <math.h>

#define Bq 8
#define Lq 2048
#define Dq 64
#define Kq 40
#define NUq 32
#define Tq 2047   // L-1 scan steps

typedef __attribute__((ext_vector_type(2))) float v2f;
typedef __attribute__((ext_vector_type(8))) float v8f;

// ---------------------------------------------------------------------------
// Kernel 1: xt = x @ W_in  (16384x64 @ 64x32), fp32 WMMA 16x16x4.
// One wave per 16x16 output tile; K=64 consumed in 16 chunks of 4.
// A layout (16x4 f32, 2 VGPRs): lanes0-15 M=0..15 hold K={0,1}; lanes16-31 K={2,3}.
// B layout (4x16 f32, 2 VGPRs): lanes0-15 hold rows K={0,1}; lanes16-31 K={2,3}.
// C/D (16x16 f32, 8 VGPRs): VGPR v: lanes0-15 -> M=v, N=lane; lanes16-31 -> M=v+8.
// ---------------------------------------------------------------------------
__global__ __launch_bounds__(32) void k_proj(const float* __restrict__ x,
                                             const float* __restrict__ Win,
                                             float* __restrict__ xt) {
  const int lane = threadIdx.x;
  const int tile = blockIdx.x;     // 0..2047
  const int tm = tile >> 1;        // M tile (16 rows of flattened B*L)
  const int tn = tile & 1;         // N tile (u 0..15 or 16..31)
  const int row0 = tm * 16;
  const int u0 = tn * 16;
  const int rIn = lane & 15;
  const int kkb = (lane >> 4) << 1;  // 0 (lanes 0-15) or 2 (lanes 16-31)

  const float* xrow = x + (size_t)(row0 + rIn) * Dq;

  v8f acc = {};
#pragma unroll
  for (int kb = 0; kb < 16; ++kb) {
    const int d0 = kb * 4;
    v2f a, bm;
    a.x = xrow[d0 + kkb];
    a.y = xrow[d0 + kkb + 1];
    bm.x = Win[(d0 + kkb) * NUq + u0 + rIn];
    bm.y = Win[(d0 + kkb + 1) * NUq + u0 + rIn];
    acc = __builtin_amdgcn_wmma_f32_16x16x4_f32(false, a, false, bm,
                                                (short)0, acc, false, false);
  }

  const int mAdd = (lane >> 4) << 3;  // 0 or 8
#pragma unroll
  for (int v = 0; v < 8; ++v) {
    xt[(size_t)(row0 + v + mAdd) * NUq + u0 + rIn] = acc[v];
  }
}

// ---------------------------------------------------------------------------
// Kernel 2: per-(b,k) delayed-reservoir scan with fused Gram accumulation.
// One wave32 per (b,k). Lane m owns: column m of Wr, column m of HtH,
// columns m and m+32 of HtX. Delay line (up to 40 states of 32 floats) in LDS.
//   H[t] = tanh(xt[b,t] + H[t-delay] @ Wr),  delay = k+1
//   for t < 2047-k:  HtH += h h^T, HtX += h x[b,t+delay]^T, s += h, xs += sum(x)
// ---------------------------------------------------------------------------
__global__ __launch_bounds__(32) void k_scan(const float* __restrict__ x,
                                             const float* __restrict__ radii,
                                             const float* __restrict__ Wres,
                                             const float* __restrict__ xt,
                                             float* __restrict__ hthW,
                                             float* __restrict__ htxW,
                                             float* __restrict__ sW,
                                             float* __restrict__ xsW) {
  const int m = threadIdx.x;
  const int bk = blockIdx.x;
  const int b = bk / Kq;
  const int k = bk % Kq;
  const int delay = k + 1;
  const int Tm = Tq - k;  // mask: t < L-1-k
  const float rad = radii[k];

  float Wc[32];
#pragma unroll
  for (int n = 0; n < 32; ++n)
    Wc[n] = Wres[(size_t)k * NUq * NUq + n * NUq + m] * rad;

  float hth[32], htx0[32], htx1[32];
#pragma unroll
  for (int n = 0; n < 32; ++n) { hth[n] = 0.f; htx0[n] = 0.f; htx1[n] = 0.f; }
  float sm = 0.f, xs = 0.f;

  __shared__ float line[Kq * NUq];
  __shared__ float red[32];

  const float* xb  = x  + (size_t)b * Lq * Dq;
  const float* xtb = xt + (size_t)b * Lq * NUq;

  int slot = 0;  // slot == t % delay; holds H[t-delay] at top of iteration t
  for (int t = 0; t < Tq; ++t) {
    float acc = xtb[t * NUq + m];
    if (t >= delay) {
#pragma unroll
      for (int n = 0; n < 32; ++n)
        acc = fmaf(line[slot * NUq + n], Wc[n], acc);
    }
    const float h = tanhf(acc);

    __syncthreads();                 // all lanes done reading old slot
    line[slot * NUq + m] = h;        // overwrite with H[t]
    __syncthreads();                 // H[t] visible to all lanes

    if (t < Tm) {
      const float* xn = xb + (size_t)(t + delay) * Dq;
      const float xa = xn[m];        // d = m
      const float xc = xn[m + 32];   // d = m+32
#pragma unroll
      for (int n = 0; n < 32; ++n) {
        const float hn = line[slot * NUq + n];
        hth[n]  = fmaf(hn, h,  hth[n]);
        htx0[n] = fmaf(hn, xa, htx0[n]);
        htx1[n] = fmaf(hn, xc, htx1[n]);
      }
      sm += h;
      xs += xa + xc;
    }
    ++slot;
    if (slot == delay) slot = 0;
  }

  float* HtHp = hthW + (size_t)bk * (NUq * NUq);
#pragma unroll
  for (int n = 0; n < 32; ++n)
    HtHp[n * 32 + m] = hth[n] + ((n == m) ? 1.0f : 0.0f);  // + REG*I

  float* HtXp = htxW + (size_t)bk * (NUq * Dq);
#pragma unroll
  for (int n = 0; n < 32; ++n) {
    HtXp[n * 64 + m]      = htx0[n];
    HtXp[n * 64 + 32 + m] = htx1[n];
  }
  sW[(size_t)bk * 32 + m] = sm;

  red[m] = xs;
  __syncthreads();
  if (m == 0) {
    float tot = 0.f;
    for (int i = 0; i < 32; ++i) tot += red[i];
    xsW[bk] = tot;
  }
}

// ---------------------------------------------------------------------------
// Kernel 3: per-(b,k) solve (HtH) W = HtX via Gauss-Jordan in LDS
// (HtH = Gram + I is SPD -> no pivoting needed). Writes W into features
// output and the collapsed f_norms contribution dot = s.W.1 - xsum.
// ---------------------------------------------------------------------------
__global__ __launch_bounds__(128) void k_solve(const float* __restrict__ hthW,
                                               const float* __restrict__ htxW,
                                               const float* __restrict__ sW,
                                               const float* __restrict__ xsW,
                                               float* __restrict__ outFeat,
                                               float* __restrict__ dotW) {
  const int tid = threadIdx.x;
  const int bk = blockIdx.x;
  __shared__ float MA[32 * 33];   // padded 32x32
  __shared__ float MB[32 * 64];
  __shared__ float rowbuf[96];
  __shared__ float fvec[32];
  __shared__ float redc[128];

  const float* A  = hthW + (size_t)bk * 1024;
  const float* Bm = htxW + (size_t)bk * 2048;
  for (int i = tid; i < 1024; i += 128) MA[(i >> 5) * 33 + (i & 31)] = A[i];
  for (int i = tid; i < 2048; i += 128) MB[i] = Bm[i];
  __syncthreads();

  for (int c = 0; c < 32; ++c) {
    const float inv = 1.0f / MA[c * 33 + c];
    if (tid < 96) {
      const float v = (tid < 32) ? MA[c * 33 + tid] : MB[c * 64 + (tid - 32)];
      rowbuf[tid] = v * inv;
    }
    if (tid < 32) fvec[tid] = MA[tid * 33 + c];
    __syncthreads();
    if (tid < 96) {
      const float rb = rowbuf[tid];
      if (tid < 32) {
        for (int rr = 0; rr < 32; ++rr) {
          const float v = MA[rr * 33 + tid];
          MA[rr * 33 + tid] = (rr == c) ? rb : (v - fvec[rr] * rb);
        }
      } else {
        const int j = tid - 32;
        for (int rr = 0; rr < 32; ++rr) {
          const float v = MB[rr * 64 + j];
          MB[rr * 64 + j] = (rr == c) ? rb : (v - fvec[rr] * rb);
        }
      }
    }
    __syncthreads();
  }

  // MB now holds W_out (32x64). features[b,k,n*64+d] = W[n][d].
  float* of = outFeat + (size_t)bk * 2048;
  const float* sv = sW + (size_t)bk * 32;
  float part = 0.f;
  for (int i = tid; i < 2048; i += 128) {
    const float w = MB[i];
    of[i] = w;
    part += sv[i >> 6] * w;
  }
  redc[tid] = part;
  __syncthreads();
  if (tid == 0) {
    float tot = 0.f;
    for (int i = 0; i < 128; ++i) tot += redc[i];
    dotW[bk] = tot - xsW[bk];
  }
}

// ---------------------------------------------------------------------------
// Kernel 4: f_norms[k] = sum_b dot[b,k]  (deterministic, no atomics)
// ---------------------------------------------------------------------------
__global__ __launch_bounds__(64) void k_fnorm(const float* __restrict__ dotW,
                                              float* __restrict__ outF) {
  const int k = threadIdx.x;
  if (k < Kq) {
    float tot = 0.f;
    for (int b = 0; b < Bq; ++b) tot += dotW[b * Kq + k];
    outF[k] = tot;
  }
}

extern "C" void kernel_launch(void* const* d_in, const int* in_sizes, int n_in,
                              void* d_out, int out_size, void* d_ws, size_t ws_size,
                              hipStream_t stream) {
  (void)in_sizes; (void)n_in; (void)out_size; (void)ws_size;
  const float* x     = (const float*)d_in[0];
  const float* radii = (const float*)d_in[1];
  const float* Win   = (const float*)d_in[2];
  const float* Wres  = (const float*)d_in[3];
  float* out = (float*)d_out;

  float* ws   = (float*)d_ws;
  float* xt   = ws;                                   // B*L*NU      = 524288
  float* hthW = xt   + (size_t)Bq * Lq * NUq;         // B*K*NU*NU   = 327680
  float* htxW = hthW + (size_t)Bq * Kq * NUq * NUq;   // B*K*NU*D    = 655360
  float* sW   = htxW + (size_t)Bq * Kq * NUq * Dq;    // B*K*NU      = 10240
  float* xsW  = sW   + (size_t)Bq * Kq * NUq;         // B*K         = 320
  float* dotW = xsW  + (size_t)Bq * Kq;               // B*K         = 320

  hipLaunchKernelGGL(k_proj, dim3((Bq * Lq / 16) * 2), dim3(32), 0, stream,
                     x, Win, xt);
  hipLaunchKernelGGL(k_scan, dim3(Bq * Kq), dim3(32), 0, stream,
                     x, radii, Wres, xt, hthW, htxW, sW, xsW);
  hipLaunchKernelGGL(k_solve, dim3(Bq * Kq), dim3(128), 0, stream,
                     hthW, htxW, sW, xsW, out, dotW);
  hipLaunchKernelGGL(k_fnorm, dim3(1), dim3(64), 0, stream,
                     dotW, out + (size_t)Bq * Kq * 2048);
}